// Model_71468255805534
// MI455X (gfx1250) — compile-verified
//
#include <hip/hip_runtime.h>
#include <math.h>

typedef float v2f __attribute__((ext_vector_type(2)));
typedef float v8f __attribute__((ext_vector_type(8)));
typedef int   v4i __attribute__((vector_size(16)));   // matches builtin param type

#define Hh    64
#define G3    192          // 3*H
#define IDIM  300
#define Bsz   256
#define Tsz   512
#define BT    (Bsz * Tsz)  // 131072
#define XS_STRIDE 308      // padded LDS row stride for x tile (conflict-free)

// ---------------------------------------------------------------------------
// gfx1250 async global->LDS copy (ASYNCcnt-tracked), 16 bytes per lane.
// Builtin signature (from hipcc diagnostic): (v4i AS1* src, v4i AS3* dst,
// int offset, int cpol). Address-space pointers produced via integer casts:
// generic LDS pointers carry the 32-bit LDS offset in their low bits; global
// generic pointers are identity-mapped.
// ---------------------------------------------------------------------------
__device__ __forceinline__ void async_copy_b128(const float* gsrc, float* lds_dst)
{
    __builtin_amdgcn_global_load_async_to_lds_b128(
        (__attribute__((address_space(1))) v4i*)(uintptr_t)gsrc,
        (__attribute__((address_space(3))) v4i*)(uint32_t)(uintptr_t)lds_dst,
        0, 0);
}

__device__ __forceinline__ void wait_async_zero()
{
    asm volatile("s_wait_asynccnt 0x0" ::: "memory");
}

// ---------------------------------------------------------------------------
// Phase 1: input projection  xp[dir][b*T+t][g] = x[b,t,:] . W_ih[dir][g,:] + b_ih
// One block per 16-row M-tile. The x tile [16 x 300] is staged once into LDS
// with async b128 loads; 12 waves then each compute N-tile w for both
// directions (24 tiles total) with V_WMMA_F32_16X16X4_F32, K = 75 chunks of 4.
// A-frag (16x4 f32): lane l: m=l&15, kh=l>>4; a[j] = X[m, k0 + j + 2*kh]
// B-frag (4x16 f32): lane l: n=l&15, kh=l>>4; b[j] = W[n, k0 + j + 2*kh]
// ---------------------------------------------------------------------------
__global__ __launch_bounds__(384) void proj_kernel(
    const float* __restrict__ x,
    const float* __restrict__ Wf, const float* __restrict__ bf,
    const float* __restrict__ Wb, const float* __restrict__ bbias,
    float* __restrict__ xpf, float* __restrict__ xpb)
{
    __shared__ float xs[16][XS_STRIDE];   // 19.7 KB staged x tile

    const int tid  = threadIdx.x;
    const int lane = tid & 31;
    const int w    = tid >> 5;            // wave = N-tile 0..11
    const int mt   = blockIdx.x;          // M-tile 0..8191

    // ---- stage x[mt*16 .. +16, 0..300) into LDS: 16 rows x 75 b128 chunks ----
    for (int i = tid; i < 16 * 75; i += 384) {
        const int r = i / 75;
        const int c = i % 75;
        async_copy_b128(x + ((size_t)(mt * 16 + r) * IDIM + c * 4), &xs[r][c * 4]);
    }
    wait_async_zero();
    __syncthreads();

    const int    mn   = lane & 15;
    const int    kh   = lane >> 4;
    const size_t brow = (size_t)(w * 16 + mn) * IDIM;
    const int    col  = w * 16 + mn;

    for (int dir = 0; dir < 2; ++dir) {
        const float* W    = dir ? Wb    : Wf;
        const float* bias = dir ? bbias : bf;
        float*       out  = dir ? xpb   : xpf;

        v8f acc = {0.f, 0.f, 0.f, 0.f, 0.f, 0.f, 0.f, 0.f};
        for (int kc = 0; kc < 75; ++kc) {
            const int k0 = kc * 4 + 2 * kh;
            v2f a, b;
            a[0] = xs[mn][k0];
            a[1] = xs[mn][k0 + 1];
            b = *(const v2f*)(W + brow + k0);
            acc = __builtin_amdgcn_wmma_f32_16x16x4_f32(
                false, a, false, b, (short)0, acc, false, false);
        }

        const float  bv    = bias[col];
        const size_t obase = (size_t)(mt * 16) * G3;
#pragma unroll
        for (int v = 0; v < 8; ++v) {
            const int row = v + 8 * kh;   // C/D: VGPR v -> M=v (lanes0-15), M=v+8 (lanes16-31)
            out[obase + (size_t)row * G3 + col] = acc[v] + bv;
        }
    }
}

// ---------------------------------------------------------------------------
// Phase 2: GRU scan. grid = (16 batch tiles, 2 directions), block = 384 = 12 waves.
// Wave w owns gate-column tile [w*16, w*16+16). W_hh staged to LDS via async
// b128; h exchanged through LDS each step. Only out[0]/out[-1] states emitted:
// feat[b][0:64]=fwd after step 1, [64:128]=bwd final, [128:192]=fwd final,
// [192:256]=bwd after step 1.
// ---------------------------------------------------------------------------
__global__ __launch_bounds__(384) void scan_kernel(
    const float* __restrict__ xpf, const float* __restrict__ xpb,
    const float* __restrict__ Whhf, const float* __restrict__ bhhf,
    const float* __restrict__ Whhb, const float* __restrict__ bhhb,
    float* __restrict__ feat)
{
    __shared__ float Wbuf[G3][Hh + 4];    // stride 68 floats: rows 16B-aligned, conflict-free
    __shared__ float hbuf[16][Hh + 4];
    __shared__ float gbuf[16][G3 + 4];

    const int tid  = threadIdx.x;
    const int lane = tid & 31;
    const int w    = tid >> 5;            // N-tile 0..11
    const int bt   = blockIdx.x;          // batch tile 0..15
    const int dir  = blockIdx.y;

    const float* xp  = dir ? xpb  : xpf;
    const float* Whh = dir ? Whhb : Whhf;
    const float* bhh = dir ? bhhb : bhhf;

    // ---- async-stage W_hh [192 x 64] into LDS: 192 rows x 16 b128 chunks ----
    for (int i = tid; i < G3 * 16; i += 384) {
        const int g = i >> 4;
        const int c = i & 15;
        async_copy_b128(Whh + (size_t)g * Hh + c * 4, &Wbuf[g][c * 4]);
    }
    for (int i = tid; i < 16 * Hh; i += 384) hbuf[i / Hh][i % Hh] = 0.f;
    wait_async_zero();
    __syncthreads();

    const int   mn  = lane & 15;
    const int   kh  = lane >> 4;
    const int   col = w * 16 + mn;        // global gate column of this lane
    const float bv  = bhh[col];

    for (int s = 0; s < Tsz; ++s) {
        // gh = h @ W_hh^T + b_hh for columns [w*16, w*16+16)
        v8f acc = {0.f, 0.f, 0.f, 0.f, 0.f, 0.f, 0.f, 0.f};
        for (int kc = 0; kc < 16; ++kc) {
            const int k0 = kc * 4 + 2 * kh;
            v2f a, b;
            a[0] = hbuf[mn][k0];
            a[1] = hbuf[mn][k0 + 1];
            b[0] = Wbuf[col][k0];
            b[1] = Wbuf[col][k0 + 1];
            acc = __builtin_amdgcn_wmma_f32_16x16x4_f32(
                false, a, false, b, (short)0, acc, false, false);
        }
#pragma unroll
        for (int v = 0; v < 8; ++v)
            gbuf[v + 8 * kh][col] = acc[v] + bv;
        __syncthreads();

        const int t = dir ? (Tsz - 1 - s) : s;

        float hnew[3];                    // 1024 elems / 384 threads -> <=3 each
        for (int idx = tid, c = 0; idx < 16 * Hh; idx += 384, ++c) {
            const int    bb  = idx >> 6;
            const int    j   = idx & 63;
            const size_t row = ((size_t)(bt * 16 + bb) * Tsz + t) * G3;
            const float xr = xp[row + j];
            const float xz = xp[row + 64 + j];
            const float xn = xp[row + 128 + j];
            const float gr = gbuf[bb][j];
            const float gz = gbuf[bb][64 + j];
            const float gn = gbuf[bb][128 + j];
            const float r  = 1.f / (1.f + expf(-(xr + gr)));
            const float z  = 1.f / (1.f + expf(-(xz + gz)));
            const float nn = tanhf(xn + r * gn);
            hnew[c] = (1.f - z) * nn + z * hbuf[bb][j];
        }
        __syncthreads();                  // all reads of hbuf/gbuf done
        for (int idx = tid, c = 0; idx < 16 * Hh; idx += 384, ++c) {
            const int bb = idx >> 6;
            const int j  = idx & 63;
            hbuf[bb][j] = hnew[c];
            if (s == 0 || s == Tsz - 1) {
                const int gbase = (s == 0) ? (dir ? 192 : 0) : (dir ? 64 : 128);
                feat[(size_t)(bt * 16 + bb) * 256 + gbase + j] = hnew[c];
            }
        }
        __syncthreads();                  // hbuf visible for next step
    }
}

// ---------------------------------------------------------------------------
// Phase 3: tiny MLP.  out[b] = W2 . leaky(W1 . feat[b] + b1) + b2
// ---------------------------------------------------------------------------
__global__ __launch_bounds__(256) void mlp_kernel(
    const float* __restrict__ feat,
    const float* __restrict__ W1, const float* __restrict__ b1,
    const float* __restrict__ W2, const float* __restrict__ b2,
    float* __restrict__ out)
{
    __shared__ float W1s[32][256];
    __shared__ float W2s[32];
    const int tid = threadIdx.x;
    for (int i = tid; i < 32 * 256; i += 256) W1s[i >> 8][i & 255] = W1[i];
    if (tid < 32) W2s[tid] = W2[tid];
    __syncthreads();

    const float* fr = feat + (size_t)tid * 256;
    float oacc = 0.f;
#pragma unroll 4
    for (int i = 0; i < 32; ++i) {
        float a = b1[i];
        for (int k = 0; k < 256; ++k) a += fr[k] * W1s[i][k];
        a = (a >= 0.f) ? a : 0.01f * a;   // LeakyReLU
        oacc += a * W2s[i];
    }
    out[tid] = oacc + b2[0];
}

// ---------------------------------------------------------------------------
extern "C" void kernel_launch(void* const* d_in, const int* in_sizes, int n_in,
                              void* d_out, int out_size, void* d_ws, size_t ws_size,
                              hipStream_t stream)
{
    (void)in_sizes; (void)n_in; (void)out_size; (void)ws_size;
    const float* x      = (const float*)d_in[0];
    const float* W_ih_f = (const float*)d_in[1];
    const float* W_hh_f = (const float*)d_in[2];
    const float* b_ih_f = (const float*)d_in[3];
    const float* b_hh_f = (const float*)d_in[4];
    const float* W_ih_b = (const float*)d_in[5];
    const float* W_hh_b = (const float*)d_in[6];
    const float* b_ih_b = (const float*)d_in[7];
    const float* b_hh_b = (const float*)d_in[8];
    const float* W1     = (const float*)d_in[9];
    const float* b1     = (const float*)d_in[10];
    const float* W2     = (const float*)d_in[11];
    const float* b2     = (const float*)d_in[12];

    float* ws   = (float*)d_ws;
    float* xpf  = ws;                                   // [BT, 192]
    float* xpb  = xpf + (size_t)BT * G3;                // [BT, 192]
    float* feat = xpb + (size_t)BT * G3;                // [256, 256]

    proj_kernel<<<dim3(BT / 16), 384, 0, stream>>>(
        x, W_ih_f, b_ih_f, W_ih_b, b_ih_b, xpf, xpb);

    scan_kernel<<<dim3(16, 2), 384, 0, stream>>>(
        xpf, xpb, W_hh_f, b_hh_f, W_hh_b, b_hh_b, feat);

    mlp_kernel<<<1, 256, 0, stream>>>(
        feat, W1, b1, W2, b2, (float*)d_out);
}